// GAT_36026185678962
// MI455X (gfx1250) — compile-verified
//
#include <hip/hip_runtime.h>
#include <math.h>

typedef __attribute__((ext_vector_type(2))) float v2f;
typedef __attribute__((ext_vector_type(4))) float v4f;
typedef __attribute__((ext_vector_type(8))) float v8f;

#define KMAX 512            // max K across layers (FIN)
#define APAD 4              // pad A rows by 4 floats -> conflict-free LDS banks

// ---------------------------------------------------------------------------
// WMMA FP32 GEMM: C[M,Nc] = A[M,K] @ B[K,Nc], row-major.
// M % 16 == 0, Nc == 64*TPW*... (block = 4 waves, wave owns TPW col tiles,
// so Nc must equal 64*TPW). K % 4 == 0, K <= KMAX.
// The 16xK A tile is staged in LDS once per block (read from HBM exactly once),
// with a +4 float row pad so the 16-lane fragment reads hit 64 distinct banks.
// Each wave runs TPW independent WMMA accumulator chains (latency hiding).
// EXEC is all-ones everywhere (exact tiling, uniform loops) as WMMA requires.
// ---------------------------------------------------------------------------
template <int TPW>
__global__ void gat_gemm_wmma(const float* __restrict__ A,
                              const float* __restrict__ B,
                              float* __restrict__ C,
                              int M, int K, int Nc) {
  __shared__ __align__(16) float As[16][KMAX + APAD];

  const int tid  = threadIdx.x;        // 128 threads = 4 waves
  const int wave = tid >> 5;
  const int lane = tid & 31;
  const int row0 = blockIdx.x * 16;

  // ---- stage A tile: 16 rows x K floats, b128 global loads -> b128 LDS stores
  const int vecs = K >> 2;             // float4 per row
  for (int i = tid; i < 16 * vecs; i += 128) {
    const int r  = i / vecs;
    const int c4 = i - r * vecs;
    v4f v = ((const v4f*)(A + (size_t)(row0 + r) * K))[c4];
    ((v4f*)&As[r][0])[c4] = v;
  }
  __syncthreads();

  const int mn = lane & 15;            // A row in tile / B col in tile
  const int kb = (lane >> 4) * 2;      // lanes 0-15: K 0,1 ; lanes 16-31: K 2,3

  v8f acc[TPW];
#pragma unroll
  for (int t = 0; t < TPW; ++t) acc[t] = (v8f){};

  for (int k = 0; k < K; k += 4) {
    v2f a;
    a.x = As[mn][k + kb];              // ds_load_b64 (bank-conflict free)
    a.y = As[mn][k + kb + 1];
#pragma unroll
    for (int t = 0; t < TPW; ++t) {
      const int c0 = (wave * TPW + t) * 16;
      v2f b;
      b.x = B[(size_t)(k + kb) * Nc + c0 + mn];
      b.y = B[(size_t)(k + kb + 1) * Nc + c0 + mn];
      acc[t] = __builtin_amdgcn_wmma_f32_16x16x4_f32(
          /*neg_a=*/false, a, /*neg_b=*/false, b,
          /*c_mod=*/(short)0, acc[t], /*reuse_a=*/false, /*reuse_b=*/false);
    }
  }

  // D layout: VGPR r -> row r (lanes 0-15) / row r+8 (lanes 16-31), col = lane&15
  const int rbase = (lane >> 4) * 8;
#pragma unroll
  for (int t = 0; t < TPW; ++t) {
    const int c0 = (wave * TPW + t) * 16;
#pragma unroll
    for (int r = 0; r < 8; ++r)
      C[(size_t)(row0 + rbase + r) * Nc + c0 + mn] = acc[t][r];
  }
}

// ---------------------------------------------------------------------------
// al_s[n,h] = sum_c h[n,h,c]*a_s[h,c] ; al_d likewise. One thread per (n,h).
// ---------------------------------------------------------------------------
__global__ void gat_attn_logits(const float* __restrict__ hfeat,
                                const float* __restrict__ a_s,
                                const float* __restrict__ a_d,
                                float* __restrict__ al_s,
                                float* __restrict__ al_d,
                                int n, int Hh, int Cc) {
  int idx = blockIdx.x * blockDim.x + threadIdx.x;
  if (idx >= n * Hh) return;
  int node = idx / Hh, h = idx - node * Hh;
  const float* hp = hfeat + (size_t)node * Hh * Cc + h * Cc;
  const float* asp = a_s + h * Cc;
  const float* adp = a_d + h * Cc;
  float ss = 0.f, sd = 0.f;
  for (int c = 0; c < Cc; ++c) {
    float v = hp[c];
    ss += v * asp[c];
    sd += v * adp[c];
  }
  al_s[idx] = ss;
  al_d[idx] = sd;
}

__global__ void gat_fill(float* __restrict__ p, int n, float v) {
  int i = blockIdx.x * blockDim.x + threadIdx.x;
  if (i < n) p[i] = v;
}

// Pass 1: e = leakyrelu(al_s[src]+al_d[dst]); store; segment-max into m[dst,h]
__global__ void gat_edge_logit_max(const int* __restrict__ ei, int E, int N, int Hh,
                                   const float* __restrict__ al_s,
                                   const float* __restrict__ al_d,
                                   float* __restrict__ ebuf,
                                   float* __restrict__ m) {
  int idx = blockIdx.x * blockDim.x + threadIdx.x;
  int tot = (E + N) * Hh;
  if (idx >= tot) return;
  int e = idx / Hh, h = idx - e * Hh;
  int src, dst;
  if (e < E) { src = ei[e]; dst = ei[E + e]; } else { src = dst = e - E; }
  float v = al_s[src * Hh + h] + al_d[dst * Hh + h];
  v = v > 0.f ? v : 0.2f * v;                    // LeakyReLU(0.2)
  ebuf[idx] = v;
  atomicMax(&m[dst * Hh + h], v);                // global_atomic_max_num_f32
}

// Pass 2: ex = exp(e - m[dst]); store; segment-sum into s[dst,h]
__global__ void gat_edge_exp_sum(const int* __restrict__ ei, int E, int N, int Hh,
                                 const float* __restrict__ m,
                                 float* __restrict__ ebuf,
                                 float* __restrict__ s) {
  int idx = blockIdx.x * blockDim.x + threadIdx.x;
  int tot = (E + N) * Hh;
  if (idx >= tot) return;
  int e = idx / Hh, h = idx - e * Hh;
  int dst = (e < E) ? ei[E + e] : e - E;
  float ex = expf(ebuf[idx] - m[dst * Hh + h]);
  ebuf[idx] = ex;
  atomicAdd(&s[dst * Hh + h], ex);
}

// Pass 3: alpha = ex / (s[dst] + 1e-16)
__global__ void gat_edge_alpha(const int* __restrict__ ei, int E, int N, int Hh,
                               const float* __restrict__ s,
                               float* __restrict__ ebuf) {
  int idx = blockIdx.x * blockDim.x + threadIdx.x;
  int tot = (E + N) * Hh;
  if (idx >= tot) return;
  int e = idx / Hh, h = idx - e * Hh;
  int dst = (e < E) ? ei[E + e] : e - E;
  ebuf[idx] = ebuf[idx] / (s[dst * Hh + h] + 1e-16f);
}

// Pass 4: out[dst, f] += h[src, f] * alpha[e, f/Cc]; one 32-lane group per edge.
__global__ void gat_edge_aggregate(const int* __restrict__ ei, int E, int N,
                                   int Hh, int Cc,
                                   const float* __restrict__ hfeat,
                                   const float* __restrict__ ebuf,
                                   float* __restrict__ outp) {
  int tid = blockIdx.x * blockDim.x + threadIdx.x;
  int e = tid >> 5;
  int lane = tid & 31;
  if (e >= E + N) return;
  int src, dst;
  if (e < E) { src = ei[e]; dst = ei[E + e]; } else { src = dst = e - E; }
  int HC = Hh * Cc;
  const float* hp = hfeat + (size_t)src * HC;
  float* op = outp + (size_t)dst * HC;
  for (int f = lane; f < HC; f += 32) {
    float a = ebuf[e * Hh + (f / Cc)];
    atomicAdd(&op[f], hp[f] * a);
  }
}

// Fused bias + LayerNorm + ELU. One wave32 per node, wave shuffle reductions.
// F in {64, 128}; F/32 <= 4 values per lane.
__global__ void gat_bias_ln_elu(const float* __restrict__ acc,
                                const float* __restrict__ bias,
                                const float* __restrict__ g,
                                const float* __restrict__ be,
                                float* __restrict__ outp, int n, int F) {
  int wave = threadIdx.x >> 5;
  int lane = threadIdx.x & 31;
  int node = blockIdx.x * (blockDim.x >> 5) + wave;
  if (node >= n) return;
  int per = F >> 5;
  float vals[4];
  float sum = 0.f;
  const float* ap = acc + (size_t)node * F;
  for (int i = 0; i < per; ++i) {
    float v = ap[i * 32 + lane] + bias[i * 32 + lane];
    vals[i] = v;
    sum += v;
  }
  for (int o = 16; o > 0; o >>= 1) sum += __shfl_xor(sum, o, 32);
  float mu = sum / (float)F;
  float vs = 0.f;
  for (int i = 0; i < per; ++i) {
    float d = vals[i] - mu;
    vs += d * d;
  }
  for (int o = 16; o > 0; o >>= 1) vs += __shfl_xor(vs, o, 32);
  float inv = rsqrtf(vs / (float)F + 1e-5f);
  float* op = outp + (size_t)node * F;
  for (int i = 0; i < per; ++i) {
    float y = g[i * 32 + lane] * (vals[i] - mu) * inv + be[i * 32 + lane];
    op[i * 32 + lane] = y > 0.f ? y : (expf(y) - 1.f);   // ELU
  }
}

// Two-stage global mean/max pool: per-block partials then 64 atomics/block.
__global__ void gat_pool_partial(const float* __restrict__ h3, int n, int F,
                                 float* __restrict__ psum,
                                 float* __restrict__ pmax) {
  int f = threadIdx.x;            // blockDim.x == F
  float s = 0.f, mx = -1e30f;
  for (int node = blockIdx.x; node < n; node += gridDim.x) {
    float v = h3[(size_t)node * F + f];
    s += v;
    mx = fmaxf(mx, v);
  }
  atomicAdd(&psum[f], s);
  atomicMax(&pmax[f], mx);
}

// z = [mean, max]; out = relu(z@cW1 + cb1) @ cW2 + cb2. Single block of 128.
__global__ void gat_classifier(const float* __restrict__ psum,
                               const float* __restrict__ pmax, int n, int F,
                               const float* __restrict__ cW1,
                               const float* __restrict__ cb1,
                               const float* __restrict__ cW2,
                               const float* __restrict__ cb2,
                               float* __restrict__ outp, int NCd) {
  __shared__ float z[128];
  __shared__ float hid[128];
  int t = threadIdx.x;            // blockDim.x == 128 == 2*F == HID
  z[t] = (t < F) ? (psum[t] / (float)n) : pmax[t - F];
  __syncthreads();
  float a = cb1[t];
  for (int i = 0; i < 2 * F; ++i) a += z[i] * cW1[i * 128 + t];
  hid[t] = fmaxf(a, 0.f);
  __syncthreads();
  if (t < NCd) {
    float o = cb2[t];
    for (int i = 0; i < 128; ++i) o += hid[i] * cW2[i * NCd + t];
    outp[t] = o;
  }
}

// ---------------------------------------------------------------------------
// Host orchestration
// ---------------------------------------------------------------------------
static inline int ceil_div(int a, int b) { return (a + b - 1) / b; }

extern "C" void kernel_launch(void* const* d_in, const int* in_sizes, int n_in,
                              void* d_out, int out_size, void* d_ws, size_t ws_size,
                              hipStream_t stream) {
  const float* x   = (const float*)d_in[0];
  const int*   ei  = (const int*)d_in[1];
  const int N   = in_sizes[2];            // batch has N entries
  const int E   = in_sizes[1] / 2;
  const int FIN = in_sizes[0] / N;        // 512
  const int HID = in_sizes[6];            // 128 (b1)
  const int OUT = in_sizes[18];           // 64  (b3)
  const int H = 4, C = HID / H;           // 4 x 32
  const int NCo = in_sizes[24];           // 2  (cb2)

  const float* W1  = (const float*)d_in[3];
  const float* as1 = (const float*)d_in[4];
  const float* ad1 = (const float*)d_in[5];
  const float* b1  = (const float*)d_in[6];
  const float* g1  = (const float*)d_in[7];
  const float* be1 = (const float*)d_in[8];
  const float* W2  = (const float*)d_in[9];
  const float* as2 = (const float*)d_in[10];
  const float* ad2 = (const float*)d_in[11];
  const float* b2  = (const float*)d_in[12];
  const float* g2  = (const float*)d_in[13];
  const float* be2 = (const float*)d_in[14];
  const float* W3  = (const float*)d_in[15];
  const float* as3 = (const float*)d_in[16];
  const float* ad3 = (const float*)d_in[17];
  const float* b3  = (const float*)d_in[18];
  const float* g3  = (const float*)d_in[19];
  const float* be3 = (const float*)d_in[20];
  const float* cW1 = (const float*)d_in[21];
  const float* cb1 = (const float*)d_in[22];
  const float* cW2 = (const float*)d_in[23];
  const float* cb2 = (const float*)d_in[24];

  // Workspace carve-up (floats)
  float* w      = (float*)d_ws;
  float* hbuf   = w;                         w += (size_t)N * HID;   // GEMM out
  float* accbuf = w;                         w += (size_t)N * HID;   // aggregation
  float* feat   = w;                         w += (size_t)N * HID;   // layer output
  float* al_s   = w;                         w += (size_t)N * H;
  float* al_d   = w;                         w += (size_t)N * H;
  float* mbuf   = w;                         w += (size_t)N * H;
  float* sbuf   = w;                         w += (size_t)N * H;
  float* ebuf   = w;                         w += (size_t)(E + N) * H;
  float* psum   = w;                         w += 64;
  float* pmax   = w;                         w += 64;
  (void)ws_size; (void)n_in;

  const int TB = 256;
  const int EH = (E + N);                    // edges incl. self-loops

  struct Layer {
    const float* in; int K; int Hh; int Cc;
    const float* W; const float* a_s; const float* a_d;
    const float* b; const float* g; const float* be; float* out;
  };
  Layer layers[3] = {
    { x,    FIN, H, C,   W1, as1, ad1, b1, g1, be1, feat },
    { feat, HID, H, C,   W2, as2, ad2, b2, g2, be2, feat },
    { feat, HID, 1, OUT, W3, as3, ad3, b3, g3, be3, feat },
  };

  for (int L = 0; L < 3; ++L) {
    const Layer& ly = layers[L];
    const int HC = ly.Hh * ly.Cc;

    // 1) h = in @ W  (WMMA fp32; one block per 16-row tile covers all HC cols)
    if (HC == 128)
      gat_gemm_wmma<2><<<N / 16, 128, 0, stream>>>(ly.in, ly.W, hbuf, N, ly.K, HC);
    else
      gat_gemm_wmma<1><<<N / 16, 128, 0, stream>>>(ly.in, ly.W, hbuf, N, ly.K, HC);

    // 2) attention logits per (node, head)
    gat_attn_logits<<<ceil_div(N * ly.Hh, TB), TB, 0, stream>>>(
        hbuf, ly.a_s, ly.a_d, al_s, al_d, N, ly.Hh, ly.Cc);

    // 3) init segment max/sum and accumulator
    gat_fill<<<ceil_div(N * ly.Hh, TB), TB, 0, stream>>>(mbuf, N * ly.Hh, -1e30f);
    gat_fill<<<ceil_div(N * ly.Hh, TB), TB, 0, stream>>>(sbuf, N * ly.Hh, 0.f);
    gat_fill<<<ceil_div(N * HC, TB), TB, 0, stream>>>(accbuf, N * HC, 0.f);

    // 4-6) edge softmax (segment max -> exp-sum -> normalize)
    int etot = EH * ly.Hh;
    gat_edge_logit_max<<<ceil_div(etot, TB), TB, 0, stream>>>(
        ei, E, N, ly.Hh, al_s, al_d, ebuf, mbuf);
    gat_edge_exp_sum<<<ceil_div(etot, TB), TB, 0, stream>>>(
        ei, E, N, ly.Hh, mbuf, ebuf, sbuf);
    gat_edge_alpha<<<ceil_div(etot, TB), TB, 0, stream>>>(
        ei, E, N, ly.Hh, sbuf, ebuf);

    // 7) weighted aggregation (32 lanes per edge)
    gat_edge_aggregate<<<ceil_div(EH * 32, TB), TB, 0, stream>>>(
        ei, E, N, ly.Hh, ly.Cc, hbuf, ebuf, accbuf);

    // 8) bias + LayerNorm + ELU (wave per node)
    gat_bias_ln_elu<<<ceil_div(N, 4), 128, 0, stream>>>(
        accbuf, ly.b, ly.g, ly.be, ly.out, N, HC);
  }

  // Pooling + classifier
  gat_fill<<<1, 128, 0, stream>>>(psum, 64, 0.f);
  gat_fill<<<1, 128, 0, stream>>>(pmax, 64, -1e30f);
  gat_pool_partial<<<256, OUT, 0, stream>>>(feat, N, OUT, psum, pmax);
  gat_classifier<<<1, 128, 0, stream>>>(psum, pmax, N, OUT,
                                        cW1, cb1, cW2, cb2,
                                        (float*)d_out, NCo);
  (void)out_size;
}